// KNNTopoLoss_12094627905840
// MI455X (gfx1250) — compile-verified
//
#include <hip/hip_runtime.h>

typedef __attribute__((ext_vector_type(2))) float v2f;
typedef __attribute__((ext_vector_type(8))) float v8f;

#define NPTS 8192
#define DIM  128
#define KNN  8
#define BITWORDS (NPTS * (NPTS / 32))   // 2,097,152 u32 = 8 MB

// ---------------- kernel 0: zero bit-adjacency + output ----------------
__global__ __launch_bounds__(256) void zero_ws_kernel(unsigned* __restrict__ bitA,
                                                      float* __restrict__ out) {
  int i = blockIdx.x * 256 + threadIdx.x;
  if (i < BITWORDS) bitA[i] = 0u;
  if (i == 0) out[0] = 0.0f;
}

// ---------------- kernel 1: row squared norms ----------------
__global__ __launch_bounds__(256) void row_sq_kernel(const float* __restrict__ Z,
                                                     float* __restrict__ sq) {
  const int wave = threadIdx.x >> 5;
  const int lane = threadIdx.x & 31;
  const int row  = blockIdx.x * 8 + wave;
  const float4* zr = (const float4*)(Z + (size_t)row * DIM);
  float4 a = zr[lane];                       // 32 lanes x 4 = 128 elements
  float s = a.x * a.x + a.y * a.y + a.z * a.z + a.w * a.w;
#pragma unroll
  for (int off = 16; off > 0; off >>= 1) s += __shfl_xor(s, off, 32);
  if (lane == 0) sq[row] = s;
}

// ---------------- top-8 helpers ----------------
__device__ __forceinline__ void topk_insert(float (&tv)[KNN], int (&ti)[KNN],
                                            float val, int idx) {
  if (val >= tv[KNN - 1]) return;            // fast reject (also rejects +inf diag)
  float v = val; int id = idx;
#pragma unroll
  for (int q = 0; q < KNN; ++q) {
    if (v < tv[q]) {
      float fv = tv[q]; int fi = ti[q];
      tv[q] = v; ti[q] = id;
      v = fv; id = fi;
    }
  }
}

__device__ __forceinline__ void topk_merge_emit(float (&tv)[KNN], int (&ti)[KNN],
                                                int row, unsigned* __restrict__ bitA,
                                                int lane) {
  for (int round = 0; round < KNN; ++round) {
    float bestv = tv[0]; int besti = ti[0]; int bestl = lane;
#pragma unroll
    for (int off = 16; off > 0; off >>= 1) {
      float ov = __shfl_xor(bestv, off, 32);
      int   oi = __shfl_xor(besti, off, 32);
      int   ol = __shfl_xor(bestl, off, 32);
      if (ov < bestv || (ov == bestv && ol < bestl)) { bestv = ov; besti = oi; bestl = ol; }
    }
    if (lane == 0) {                          // set bits (row,c) and (c,row)
      int c = besti;
      atomicOr(&bitA[((unsigned)row << 8) + (c >> 5)], 1u << (c & 31));
      atomicOr(&bitA[((unsigned)c << 8) + (row >> 5)], 1u << (row & 31));
    }
    if (lane == bestl) {                      // winner pops its smallest (const-index shift)
#pragma unroll
      for (int q = 0; q < KNN - 1; ++q) { tv[q] = tv[q + 1]; ti[q] = ti[q + 1]; }
      tv[KNN - 1] = __builtin_inff();
    }
  }
}

// ---------------- kernel 2: WMMA Gram tiles + per-row top-8 + edge bits ----------------
__global__ __launch_bounds__(256) void knn_main_kernel(const float* __restrict__ Z,
                                                       const float* __restrict__ sq,
                                                       unsigned* __restrict__ bitA) {
  __shared__ float Alds[16][DIM];   // 8 KB: 16-row strip of Z
  __shared__ float Slds[16][128];   // 8 KB: scores for 16 rows x 128 cols

  const int tid  = threadIdx.x;
  const int wave = tid >> 5;
  const int lane = tid & 31;
  const int row0 = blockIdx.x << 4;

  // Load the 16x128 A strip (2048 floats, 8 per thread)
  {
    const float4* src = (const float4*)(Z + (size_t)row0 * DIM);
    float4* dst = (float4*)(&Alds[0][0]);
    dst[tid]       = src[tid];
    dst[tid + 256] = src[tid + 256];
  }
  __syncthreads();

  // Preload A fragments per the 16x4 f32 layout:
  // lanes 0-15: M=lane, K={k0,k0+1}; lanes 16-31: M=lane-16, K={k0+2,k0+3}
  v2f afrag[32];
  {
    const int m = lane & 15;
    const int ksel = (lane >> 4) << 1;
#pragma unroll
    for (int k = 0; k < 32; ++k)
      afrag[k] = *(const v2f*)(&Alds[m][k * 4 + ksel]);
  }

  // Two rows of running top-8 per wave (rows 2w, 2w+1 of the strip)
  float tvA[KNN], tvB[KNN]; int tiA[KNN], tiB[KNN];
#pragma unroll
  for (int q = 0; q < KNN; ++q) {
    tvA[q] = __builtin_inff(); tvB[q] = __builtin_inff();
    tiA[q] = 0; tiB[q] = 0;
  }

  for (int tile = 0; tile < NPTS / 128; ++tile) {
    const int c0 = tile * 128;
    const int cw = c0 + wave * 16;           // this wave's 16-column sub-tile

    // ---- 16x16 dot tile via 32x v_wmma_f32_16x16x4_f32 ----
    // Phase 1: batch-load ALL 32 B fragments into distinct registers so the
    // 32 global_load_b64s pipeline (loadcnt counts down) instead of
    // load->wait(0)->wmma serialization.
    v8f acc = {};
    {
      const int n = lane & 15;
      const int ksel = (lane >> 4) << 1;
      const float* bptr = Z + (size_t)(cw + n) * DIM + ksel;  // L2-resident
      v2f bfrag[32];
#pragma unroll
      for (int k = 0; k < 32; ++k)
        bfrag[k] = *(const v2f*)(bptr + k * 4);
      // Phase 2: dense WMMA accumulation chain
#pragma unroll
      for (int k = 0; k < 32; ++k)
        acc = __builtin_amdgcn_wmma_f32_16x16x4_f32(
            false, afrag[k], false, bfrag[k], (short)0, acc, false, false);
    }

    // ---- scores s = sq[col] - 2*dot (row-constant sq[i] dropped; +inf diag) ----
    {
      const int n = lane & 15;
      const int gcol = cw + n;
      const float sqc = sq[gcol];
      const int mbase = (lane >> 4) << 3;    // lanes 0-15 -> M 0..7; 16-31 -> M 8..15
#pragma unroll
      for (int v = 0; v < 8; ++v) {
        const int M = mbase + v;
        float s = __builtin_fmaf(-2.0f, acc[v], sqc);
        if (row0 + M == gcol) s = __builtin_inff();
        Slds[M][wave * 16 + n] = s;
      }
    }
    __syncthreads();

    // ---- selection: wave owns rows 2w,2w+1; each lane scans 4 columns ----
#pragma unroll
    for (int ch = 0; ch < 4; ++ch) {
      const int cl = lane + ch * 32;
      topk_insert(tvA, tiA, Slds[2 * wave][cl],     c0 + cl);
      topk_insert(tvB, tiB, Slds[2 * wave + 1][cl], c0 + cl);
    }
    __syncthreads();
  }

  topk_merge_emit(tvA, tiA, row0 + 2 * wave,     bitA, lane);
  topk_merge_emit(tvB, tiB, row0 + 2 * wave + 1, bitA, lane);
}

// ---------------- kernel 3: fused BCE reduction over target_adj ----------------
__global__ __launch_bounds__(256) void loss_sum_kernel(const float* __restrict__ T,
                                                       const unsigned* __restrict__ bitA,
                                                       float* __restrict__ out) {
  const long long total4 = (long long)NPTS * NPTS / 4;
  const int tid  = threadIdx.x;
  const int wave = tid >> 5;
  const int lane = tid & 31;

  float s = 0.0f;
  for (long long idx = (long long)blockIdx.x * 256 + tid; idx < total4;
       idx += (long long)gridDim.x * 256) {
    const long long f = idx * 4;             // flat element index, 4-aligned
    const float4 t = ((const float4*)T)[idx];
    const unsigned w = bitA[f >> 5];
    const int sh = (int)(f & 31);            // sh..sh+3 within one word
    s += ((w >> (sh + 0)) & 1u) ? (1.0f - t.x) : t.x;
    s += ((w >> (sh + 1)) & 1u) ? (1.0f - t.y) : t.y;
    s += ((w >> (sh + 2)) & 1u) ? (1.0f - t.z) : t.z;
    s += ((w >> (sh + 3)) & 1u) ? (1.0f - t.w) : t.w;
  }
#pragma unroll
  for (int off = 16; off > 0; off >>= 1) s += __shfl_xor(s, off, 32);

  __shared__ float wsum[8];
  if (lane == 0) wsum[wave] = s;
  __syncthreads();
  if (tid == 0) {
    float b = 0.0f;
#pragma unroll
    for (int i = 0; i < 8; ++i) b += wsum[i];
    const float scale = 100.0f / ((float)NPTS * (float)NPTS);
    atomicAdd(out, b * scale);
  }
}

// ---------------- launch ----------------
extern "C" void kernel_launch(void* const* d_in, const int* in_sizes, int n_in,
                              void* d_out, int out_size, void* d_ws, size_t ws_size,
                              hipStream_t stream) {
  const float* Z = (const float*)d_in[0];        // [8192,128] f32
  const float* T = (const float*)d_in[1];        // [8192,8192] f32
  float* out = (float*)d_out;                    // scalar f32

  unsigned* bitA = (unsigned*)d_ws;                                   // 8 MB bit adjacency
  float* sq = (float*)((char*)d_ws + (size_t)BITWORDS * sizeof(unsigned)); // 32 KB norms

  zero_ws_kernel<<<(BITWORDS + 255) / 256, 256, 0, stream>>>(bitA, out);
  row_sq_kernel<<<NPTS / 8, 256, 0, stream>>>(Z, sq);
  knn_main_kernel<<<NPTS / 16, 256, 0, stream>>>(Z, sq, bitA);
  loss_sum_kernel<<<8192, 256, 0, stream>>>(T, bitA, out);
}